// CenterLoss_71829033058385
// MI455X (gfx1250) — compile-verified
//
#include <hip/hip_runtime.h>

// CenterLoss, MI455X / gfx1250, wave32.
//
// distmat*mask is one-hot per row -> loss reduces to gathered per-row squared
// distances plus a closed-form (C-1)*1e-12 clamp constant. Total traffic
// ~1.5MB -> launch-latency bound; single fused pass in f32.
//
// Per block (1 wave, 16 rows): diag(X*G^T) and diag(Y*G^T) via
// V_WMMA_F32_16X16X4_F32 chains (K=128 in steps of 4); squared norms ride the
// same float2 loads. Deterministic 2-kernel reduction (no float atomics).

typedef __attribute__((ext_vector_type(2))) float v2f;
typedef __attribute__((ext_vector_type(8))) float v8f;

#define D_DIM 128
#define TILE 16

__global__ __launch_bounds__(32)
void center_loss_partial(const float* __restrict__ x,
                         const float* __restrict__ y,
                         const int*   __restrict__ labels,
                         const float* __restrict__ centers,
                         float*       __restrict__ partials)
{
    const int lane  = threadIdx.x;           // 0..31, wave32
    const int m     = lane & 15;             // row within 16-row tile
    const int khalf = (lane >> 4) * 2;       // K sub-pair: lanes 0-15 -> K0,K1; 16-31 -> K2,K3
    const int row   = blockIdx.x * TILE + m; // global batch row

    const float* xr = x + (size_t)row * D_DIM;
    const float* yr = y + (size_t)row * D_DIM;
    const int    lb = labels[row];           // lanes l and l+16 read the same label (uniform-ish gather)
    const float* gr = centers + (size_t)lb * D_DIM;

    v8f accX = {};            // X_tile (16xD) * G_tile^T (Dx16), f32 accumulate
    v8f accY = {};
    float px = 0.f, py = 0.f, pg = 0.f;   // per-lane partial squared norms

    // A-matrix 16x4 f32 layout == B-matrix 4x16 layout for row-major tiles:
    // lane l supplies 2 consecutive K values of row (l&15) at offset khalf.
#pragma unroll 4
    for (int k0 = 0; k0 < D_DIM; k0 += 4) {
        v2f a  = *(const v2f*)(xr + k0 + khalf);
        v2f ay = *(const v2f*)(yr + k0 + khalf);
        v2f b  = *(const v2f*)(gr + k0 + khalf);
        px += a.x  * a.x  + a.y  * a.y;
        py += ay.x * ay.x + ay.y * ay.y;
        pg += b.x  * b.x  + b.y  * b.y;
        // 8-arg form: (neg_a, A, neg_b, B, c_mod, C, reuse_a, reuse_b)
        accX = __builtin_amdgcn_wmma_f32_16x16x4_f32(false, a,  false, b, (short)0, accX, false, false);
        accY = __builtin_amdgcn_wmma_f32_16x16x4_f32(false, ay, false, b, (short)0, accY, false, false);
    }

    // Full-row norms: lane m + lane m+16 cover K=0..127 of row m.
    float nx = px + __shfl_xor(px, 16, 32);
    float ny = py + __shfl_xor(py, 16, 32);
    float nc = pg + __shfl_xor(pg, 16, 32);

    // Diagonal of C/D layout: M=0..7 at lanes 0..7 (vgpr==lane),
    // M=8..15 at lanes 24..31 (vgpr==lane-24).
    float dX = 0.f, dY = 0.f;
#pragma unroll
    for (int i = 0; i < 8; ++i) {
        bool sel = (lane == i) || (lane == 24 + i);
        dX = sel ? accX[i] : dX;
        dY = sel ? accY[i] : dY;
    }
    dX += __shfl_xor(dX, 16, 32);   // lane m (0..15) now holds x_m . c_m
    dY += __shfl_xor(dY, 16, 32);

    float contrib = 0.f;
    if (lane < TILE) {
        float distX = nx + nc - 2.0f * dX;   // expanded form, faithful to reference
        float distY = ny + nc - 2.0f * dY;
        distX = fminf(fmaxf(distX, 1e-12f), 1e12f);
        distY = fminf(fmaxf(distY, 1e-12f), 1e12f);
        contrib = distX + distY;
    }
    // Deterministic wave32 shuffle-tree reduction.
#pragma unroll
    for (int s = 16; s >= 1; s >>= 1)
        contrib += __shfl_xor(contrib, s, 32);
    if (lane == 0) partials[blockIdx.x] = contrib;
}

__global__ __launch_bounds__(32)
void center_loss_final(const float* __restrict__ partials,
                       float* __restrict__ out,
                       int nparts, float invB, float maskconst)
{
    const int lane = threadIdx.x;
    float s = 0.f;
    for (int i = lane; i < nparts; i += 32)  // fixed per-lane order -> deterministic
        s += partials[i];
#pragma unroll
    for (int sh = 16; sh >= 1; sh >>= 1)
        s += __shfl_xor(s, sh, 32);
    if (lane == 0)
        out[0] = 0.01f * (s * invB + maskconst);
}

extern "C" void kernel_launch(void* const* d_in, const int* in_sizes, int n_in,
                              void* d_out, int out_size, void* d_ws, size_t ws_size,
                              hipStream_t stream) {
    const float* x       = (const float*)d_in[0];
    const float* y       = (const float*)d_in[1];
    const int*   labels  = (const int*)  d_in[2];
    const float* centers = (const float*)d_in[3];

    const int B = in_sizes[2];            // 1024
    const int C = in_sizes[3] / D_DIM;    // 100000
    const int nblocks = B / TILE;         // 64

    float* partials = (float*)d_ws;

    center_loss_partial<<<nblocks, 32, 0, stream>>>(x, y, labels, centers, partials);

    // Each of B rows contributes (C-1) clamped zeros of 1e-12 in BOTH masked
    // sums; after /B that is 2*(C-1)*1e-12 added once.
    const float maskconst = 2.0f * (float)(C - 1) * 1e-12f;
    center_loss_final<<<1, 32, 0, stream>>>(partials, (float*)d_out,
                                            nblocks, 1.0f / (float)B, maskconst);
}